// SurvivalGraphArch_17471926960175
// MI455X (gfx1250) — compile-verified
//
#include <hip/hip_runtime.h>
#include <hip/hip_bf16.h>

typedef float v2f __attribute__((ext_vector_type(2)));
typedef float v8f __attribute__((ext_vector_type(8)));

#if defined(__HIP_DEVICE_COMPILE__) && __has_builtin(__builtin_amdgcn_global_load_async_to_lds_b128)
#define GEMM_ASYNC 1
typedef int v4i_vs __attribute__((vector_size(16)));                 // matches builtin param
typedef __attribute__((address_space(1))) v4i_vs as1_v4i;            // global
typedef __attribute__((address_space(3))) v4i_vs as3_v4i;            // LDS
#endif

// ============================================================================
// WMMA f32 GEMM:  Y[N,M] = act(X[N,K] @ W[K,M] + bias[M])
// W row-major [K,M] (din x dout, as in the reference _lin()).
// Block: 256 threads = 8 waves. Block tile: 128 rows x 64 cols.
// Each wave: 16 rows x 64 cols (4 accumulators). K staged through LDS, KC=32.
// X panel staged with GLOBAL_LOAD_ASYNC_TO_LDS_B128 when available (ASYNCcnt),
// W panel staged transposed so every WMMA fragment is one contiguous b64 load.
// act: 0=none 1=relu 2=tanh 3=sigmoid
// Requires: K % 32 == 0, M % 64 == 0. Rows beyond N may be read (A-row -> C-row
// 1:1, out-of-range C rows are not stored; callers guarantee readable slack).
// ============================================================================
__global__ __launch_bounds__(256)
void gemm_wmma_kernel(const float* __restrict__ X, const float* __restrict__ W,
                      const float* __restrict__ bias, float* __restrict__ Y,
                      int N, int K, int M, int act)
{
    const int KC = 32;
    __shared__ float Xs[128][KC + 4];       // stride 36 floats: 16B-aligned rows, conflict-free
    __shared__ float WsT[64][KC + 4];

    const int tid  = threadIdx.x;
    const int wave = tid >> 5;
    const int lane = tid & 31;
    const int rowBase = blockIdx.x * 128;
    const int colBase = blockIdx.y * 64;

    v8f acc[4];
    #pragma unroll
    for (int j = 0; j < 4; ++j)
        #pragma unroll
        for (int i = 0; i < 8; ++i) acc[j][i] = 0.0f;

    const int m    = lane & 15;
    const int koff = (lane >> 4) * 2;       // lanes 0-15: K pair {0,1}; lanes 16-31: {2,3}

    for (int k0 = 0; k0 < K; k0 += KC) {
        // ---- stage X panel: 128 rows x 32 floats ----
#ifdef GEMM_ASYNC
        {
            int r = tid >> 1;               // 0..127
            int c = (tid & 1) * 16;         // 0 or 16
            const float* gp = X + (size_t)(rowBase + r) * K + (k0 + c);
            __builtin_amdgcn_global_load_async_to_lds_b128(
                (as1_v4i*)gp, (as3_v4i*)&Xs[r][c], 0, 0);
        }
#else
        for (int i = tid; i < 128 * KC; i += 256) {
            int r = i / KC, c = i % KC;
            int gr = rowBase + r;
            Xs[r][c] = (gr < N) ? X[(size_t)gr * K + (k0 + c)] : 0.0f;
        }
#endif
        // ---- stage W panel transposed: WsT[col][k] ----
        for (int i = tid; i < KC * 64; i += 256) {
            int r = i >> 6, c = i & 63;     // r = k-row, c = col
            WsT[c][r] = W[(size_t)(k0 + r) * M + (colBase + c)];
        }
        // prefetch next W panel (global_prefetch_b8)
        if (k0 + KC < K)
            __builtin_prefetch(&W[(size_t)(k0 + KC + (tid >> 6)) * M + (colBase + (tid & 63))], 0, 0);

#ifdef GEMM_ASYNC
#if __has_builtin(__builtin_amdgcn_s_wait_asynccnt)
        __builtin_amdgcn_s_wait_asynccnt(0);
#else
        asm volatile("s_wait_asynccnt 0x0" ::: "memory");
#endif
#endif
        __syncthreads();

        const float* arow = &Xs[wave * 16 + m][0];
        #pragma unroll
        for (int kk = 0; kk < KC; kk += 4) {
            v2f a = *(const v2f*)&arow[kk + koff];
            #pragma unroll
            for (int j = 0; j < 4; ++j) {
                v2f b = *(const v2f*)&WsT[j * 16 + m][kk + koff];
                acc[j] = __builtin_amdgcn_wmma_f32_16x16x4_f32(
                    false, a, false, b, (short)0, acc[j], false, false);
            }
        }
        __syncthreads();
    }

    // C/D layout: VGPR v -> row = v + 8*(lane>=16), col = lane&15
    const int mrow = (lane >> 4) * 8;
    const int ncol = lane & 15;
    #pragma unroll
    for (int j = 0; j < 4; ++j) {
        int col = colBase + j * 16 + ncol;
        float bv = bias ? bias[col] : 0.0f;
        #pragma unroll
        for (int v = 0; v < 8; ++v) {
            int row = rowBase + wave * 16 + v + mrow;
            if (row < N) {
                float val = acc[j][v] + bv;
                if (act == 1)      val = fmaxf(val, 0.0f);
                else if (act == 2) val = tanhf(val);
                else if (act == 3) val = 1.0f / (1.0f + __expf(-val));
                Y[(size_t)row * M + col] = val;
            }
        }
    }
}

// ============================================================================
// Edge aggregation: xa[dst] += x[src] for valid edges (xa pre-initialized = x)
// ============================================================================
__global__ void agg_edges_kernel(const float* __restrict__ x, float* __restrict__ xa,
                                 const int* __restrict__ src, const int* __restrict__ dst,
                                 const int* __restrict__ valid, int E, int D)
{
    int e = blockIdx.x;
    if (e >= E) return;
    if (valid && !valid[e]) return;
    int s = src[e], d = dst[e];
    const float* xs = x + (size_t)s * D;
    float* xd = xa + (size_t)d * D;
    for (int i = threadIdx.x; i < D; i += blockDim.x) atomicAdd(&xd[i], xs[i]);
}

// ============================================================================
// BatchNorm statistics: one block per column
// ============================================================================
__global__ __launch_bounds__(256)
void bn_stats_kernel(const float* __restrict__ H, int N, int D,
                     float* __restrict__ mu, float* __restrict__ rstd)
{
    __shared__ float ss[256], sq[256];
    int c = blockIdx.x;
    float s = 0.0f, q = 0.0f;
    for (int r = threadIdx.x; r < N; r += 256) {
        float v = H[(size_t)r * D + c];
        s += v; q += v * v;
    }
    ss[threadIdx.x] = s; sq[threadIdx.x] = q;
    __syncthreads();
    for (int o = 128; o > 0; o >>= 1) {
        if (threadIdx.x < o) { ss[threadIdx.x] += ss[threadIdx.x + o]; sq[threadIdx.x] += sq[threadIdx.x + o]; }
        __syncthreads();
    }
    if (threadIdx.x == 0) {
        float mn  = ss[0] / (float)N;
        float var = sq[0] / (float)N - mn * mn;
        mu[c]   = mn;
        rstd[c] = rsqrtf(var + 1e-5f);
    }
}

__global__ void bn_apply_kernel(float* H, long long total, int D,
                                const float* __restrict__ mu, const float* __restrict__ rstd,
                                const float* __restrict__ g, const float* __restrict__ bt)
{
    long long i = (long long)blockIdx.x * blockDim.x + threadIdx.x;
    if (i >= total) return;
    int c = (int)(i % D);
    float v = (H[i] - mu[c]) * rstd[c] * g[c] + bt[c];
    H[i] = fmaxf(v, 0.0f);
}

// ============================================================================
// TopK pooling pieces
// ============================================================================
__global__ __launch_bounds__(256)
void wnorm_kernel(const float* __restrict__ w, int D, float* __restrict__ invnorm)
{
    __shared__ float ss[256];
    float s = 0.0f;
    for (int i = threadIdx.x; i < D; i += 256) { float v = w[i]; s += v * v; }
    ss[threadIdx.x] = s; __syncthreads();
    for (int o = 128; o > 0; o >>= 1) {
        if (threadIdx.x < o) ss[threadIdx.x] += ss[threadIdx.x + o];
        __syncthreads();
    }
    if (threadIdx.x == 0) *invnorm = rsqrtf(ss[0]);
}

// wave-per-row dot product: score = tanh((x . w) * invnorm)
__global__ __launch_bounds__(256)
void score_kernel(const float* __restrict__ X, int N, int D,
                  const float* __restrict__ w, const float* __restrict__ invnorm,
                  float* __restrict__ score)
{
    int row  = (int)((blockIdx.x * blockDim.x + threadIdx.x) >> 5);
    int lane = threadIdx.x & 31;
    if (row >= N) return;
    const float* x = X + (size_t)row * D;
    float s = 0.0f;
    for (int i = lane; i < D; i += 32) s += x[i] * w[i];
    #pragma unroll
    for (int o = 16; o > 0; o >>= 1) s += __shfl_xor(s, o, 32);
    if (lane == 0) score[row] = tanhf(s * (*invnorm));
}

__device__ __forceinline__ unsigned f2ord(float f) {
    unsigned u = __float_as_uint(f);
    return (u >> 31) ? ~u : (u | 0x80000000u);
}
__device__ __forceinline__ float ord2f(unsigned k) {
    unsigned u = (k & 0x80000000u) ? (k ^ 0x80000000u) : ~k;
    return __uint_as_float(u);
}

// Exact per-graph top-k: bitonic sort of (score,idx) pairs in dynamic LDS.
// One block per graph. P = npg padded to power of two; ties prefer lower index.
__global__ __launch_bounds__(256)
void topk_kernel(const float* __restrict__ score, int npg, int P, int k,
                 int* __restrict__ perm, float* __restrict__ vals, int* __restrict__ mapping)
{
    extern __shared__ unsigned long long s[];
    int b = blockIdx.x, tid = threadIdx.x;
    for (int i = tid; i < P; i += 256) {
        unsigned long long v = 0ull;
        if (i < npg) {
            unsigned key = f2ord(score[b * npg + i]);
            v = ((unsigned long long)key << 32) | (unsigned)(0xFFFFFFFFu - (unsigned)i);
            mapping[b * npg + i] = -1;
        }
        s[i] = v;
    }
    __syncthreads();
    for (int kk = 2; kk <= P; kk <<= 1) {
        for (int j = kk >> 1; j > 0; j >>= 1) {
            for (int i = tid; i < P; i += 256) {
                int ij = i ^ j;
                if (ij > i) {
                    bool up = (i & kk) != 0;        // ascending segment
                    unsigned long long a = s[i], c = s[ij];
                    bool sw = up ? (c < a) : (a < c);
                    if (sw) { s[i] = c; s[ij] = a; }
                }
            }
            __syncthreads();
        }
    }
    for (int j = tid; j < k; j += 256) {
        unsigned long long v = s[j];
        int idx = (int)(0xFFFFFFFFu - (unsigned)(v & 0xFFFFFFFFu));
        int nid = b * k + j;
        perm[nid] = b * npg + idx;
        vals[nid] = ord2f((unsigned)(v >> 32));
        mapping[b * npg + idx] = nid;
    }
}

__global__ void gather_kernel(const float* __restrict__ X, float* __restrict__ Y,
                              const int* __restrict__ perm, const float* __restrict__ vals,
                              int rows, int D)
{
    int j = blockIdx.x;
    if (j >= rows) return;
    const float* xs = X + (size_t)perm[j] * D;
    float v = vals[j];
    float* yd = Y + (size_t)j * D;
    for (int i = threadIdx.x; i < D; i += blockDim.x) yd[i] = xs[i] * v;
}

__global__ void remap_kernel(const int* srcIn, const int* dstIn, const int* validIn,
                             int* srcOut, int* dstOut, int* validOut,
                             const int* mapping, int E)
{
    int e = blockIdx.x * blockDim.x + threadIdx.x;
    if (e >= E) return;
    int s0 = srcIn[e], d0 = dstIn[e];
    int vv = validIn ? validIn[e] : 1;
    int s = mapping[s0], d = mapping[d0];
    int v = (vv && s >= 0 && d >= 0) ? 1 : 0;
    srcOut[e] = s < 0 ? 0 : s;
    dstOut[e] = d < 0 ? 0 : d;
    validOut[e] = v;
}

// ============================================================================
// Gated attention aggregation
// ============================================================================
// wave-per-row: out = sum_i a[i]*b[i]*Wc[2i] + bc[0]   (loc channel only)
__global__ __launch_bounds__(256)
void gate_kernel(const float* __restrict__ A, const float* __restrict__ Bm,
                 const float* __restrict__ Wc, const float* __restrict__ bc,
                 int N, int D, float* __restrict__ out)
{
    int row  = (int)((blockIdx.x * blockDim.x + threadIdx.x) >> 5);
    int lane = threadIdx.x & 31;
    if (row >= N) return;
    const float* a = A + (size_t)row * D;
    const float* b = Bm + (size_t)row * D;
    float acc = 0.0f;
    for (int i = lane; i < D; i += 32) acc += a[i] * b[i] * Wc[i * 2 + 0];
    #pragma unroll
    for (int o = 16; o > 0; o >>= 1) acc += __shfl_xor(acc, o, 32);
    if (lane == 0) out[row] = acc + bc[0];
}

__global__ __launch_bounds__(256)
void softmax_kernel(const float* __restrict__ s, int k, float* __restrict__ w)
{
    __shared__ float red[256];
    int b = blockIdx.x;
    const float* sb = s + (size_t)b * k;
    float m = -1e30f;
    for (int i = threadIdx.x; i < k; i += 256) m = fmaxf(m, sb[i]);
    red[threadIdx.x] = m; __syncthreads();
    for (int o = 128; o > 0; o >>= 1) {
        if (threadIdx.x < o) red[threadIdx.x] = fmaxf(red[threadIdx.x], red[threadIdx.x + o]);
        __syncthreads();
    }
    m = red[0]; __syncthreads();
    float sum = 0.0f;
    for (int i = threadIdx.x; i < k; i += 256) sum += __expf(sb[i] - m);
    red[threadIdx.x] = sum; __syncthreads();
    for (int o = 128; o > 0; o >>= 1) {
        if (threadIdx.x < o) red[threadIdx.x] += red[threadIdx.x + o];
        __syncthreads();
    }
    float inv = 1.0f / red[0];
    for (int i = threadIdx.x; i < k; i += 256) w[(size_t)b * k + i] = __expf(sb[i] - m) * inv;
}

__global__ __launch_bounds__(256)
void pool_agg_kernel(const float* __restrict__ X, const float* __restrict__ w,
                     int k, int D, float* __restrict__ out)
{
    int b = blockIdx.x;
    int c = blockIdx.y * 256 + threadIdx.x;
    if (c >= D) return;
    float acc = 0.0f;
    for (int j = 0; j < k; ++j)
        acc += w[(size_t)b * k + j] * X[(size_t)(b * k + j) * D + c];
    out[(size_t)b * D + c] = acc;
}

__global__ void cls_kernel(const float* __restrict__ pr, const float* __restrict__ pp,
                           const float* __restrict__ W, const float* __restrict__ bias,
                           float* __restrict__ out)
{
    int t = threadIdx.x;
    if (t >= 16) return;
    int b = t >> 2, c = t & 3;
    float acc = bias[c];
    for (int d = 0; d < 512; ++d) acc += pr[b * 512 + d] * W[d * 4 + c];
    for (int d = 0; d < 512; ++d) acc += pp[b * 512 + d] * W[(512 + d) * 4 + c];
    out[b * 4 + c] = acc;
}

// ============================================================================
// Host orchestration
// ============================================================================
static inline void launch_gemm(const float* X, const float* W, const float* bias, float* Y,
                               int N, int K, int M, int act, hipStream_t s)
{
    dim3 g((N + 127) / 128, M / 64);
    gemm_wmma_kernel<<<g, 256, 0, s>>>(X, W, bias, Y, N, K, M, act);
}

extern "C" void kernel_launch(void* const* d_in, const int* in_sizes, int n_in,
                              void* d_out, int out_size, void* d_ws, size_t ws_size,
                              hipStream_t stream)
{
    (void)in_sizes; (void)n_in; (void)out_size; (void)ws_size;
    const int B = 4, D = 512;
    const int NPG_R = 2048, NPG_P = 8192, DEG = 8;
    const int DF_R = 128, DF_P = 768;
    const int E_R = B * NPG_R * DEG, E_P = B * NPG_P * DEG;
    const int K_R1 = 1434, K_R2 = 1004, K_P1 = 1639, K_P2 = 328;
    const int NMAX = B * NPG_P;          // 32768
    const int KMAX1 = B * K_P1;          // 6556 (max pooled rows after pool1)
    const int GMAX = B * K_R1;           // >= max gate rows (B*k2 max = 4016)

    const float* x_r    = (const float*)d_in[0];
    const float* x_p    = (const float*)d_in[1];
    const int*   edge_r = (const int*)d_in[2];
    const int*   edge_p = (const int*)d_in[3];
    const float* Pm[36];
    for (int i = 4; i < 36; ++i) Pm[i] = (const float*)d_in[i];

    // Workspace bump allocator (~149.5 MB needed)
    char* wp = (char*)d_ws;
    auto alloc = [&](size_t bytes) -> void* {
        void* p = (void*)wp;
        wp += (bytes + 255) & ~(size_t)255;
        return p;
    };
    float* buf0    = (float*)alloc((size_t)NMAX * D * 4);
    float* buf1    = (float*)alloc((size_t)NMAX * D * 4);
    float* mu      = (float*)alloc(D * 4);
    float* rstd    = (float*)alloc(D * 4);
    float* scores  = (float*)alloc((size_t)NMAX * 4);
    int*   mapping = (int*)  alloc((size_t)NMAX * 4);
    int*   perm    = (int*)  alloc((size_t)KMAX1 * 4);
    float* vals    = (float*)alloc((size_t)KMAX1 * 4);
    int*   src2    = (int*)  alloc((size_t)E_P * 4);
    int*   dst2    = (int*)  alloc((size_t)E_P * 4);
    int*   valid2  = (int*)  alloc((size_t)E_P * 4);
    float* abuf    = (float*)alloc((size_t)GMAX * 256 * 4);
    float* bbuf    = (float*)alloc((size_t)GMAX * 256 * 4);
    float* gateS   = (float*)alloc((size_t)GMAX * 4);
    float* wgt     = (float*)alloc((size_t)GMAX * 4);
    float* pooled_r= (float*)alloc((size_t)B * D * 4);
    float* pooled_p= (float*)alloc((size_t)B * D * 4);
    float* invn    = (float*)alloc(256);

    auto run_branch = [&](const float* X0, int DF, const int* edges, int E,
                          int NPG, int k1, int k2, int p2pow,
                          const float* encW, const float* encb,
                          const float* g1W, const float* g1b, const float* g1g, const float* g1bt,
                          const float* p1w,
                          const float* g2W, const float* g2b, const float* g2g, const float* g2bt,
                          const float* p2w,
                          float* pooled)
    {
        const int N0 = B * NPG;
        const int* esrc = edges;
        const int* edst = edges + E;

        // ---- encoder: buf0 = relu(X0 @ encW + encb) ----
        launch_gemm(X0, encW, encb, buf0, N0, DF, D, 1, stream);

        // ---- GIN1: xa = x + sum_{edges} x[src] ----
        (void)hipMemcpyAsync(buf1, buf0, (size_t)N0 * D * 4, hipMemcpyDeviceToDevice, stream);
        agg_edges_kernel<<<E, 256, 0, stream>>>(buf0, buf1, esrc, edst, nullptr, E, D);
        launch_gemm(buf1, g1W, g1b, buf0, N0, D, D, 0, stream);
        bn_stats_kernel<<<D, 256, 0, stream>>>(buf0, N0, D, mu, rstd);
        {
            long long tot = (long long)N0 * D;
            bn_apply_kernel<<<(unsigned)((tot + 255) / 256), 256, 0, stream>>>(buf0, tot, D, mu, rstd, g1g, g1bt);
        }

        // ---- TopK pool 1 ----
        wnorm_kernel<<<1, 256, 0, stream>>>(p1w, D, invn);
        score_kernel<<<(N0 * 32 + 255) / 256, 256, 0, stream>>>(buf0, N0, D, p1w, invn, scores);
        topk_kernel<<<B, 256, (size_t)NPG * 8, stream>>>(scores, NPG, NPG, k1, perm, vals, mapping);
        int N1 = B * k1;
        gather_kernel<<<N1, 256, 0, stream>>>(buf0, buf1, perm, vals, N1, D);
        remap_kernel<<<(E + 255) / 256, 256, 0, stream>>>(esrc, edst, nullptr,
                                                          src2, dst2, valid2, mapping, E);

        // ---- GIN2 ----
        (void)hipMemcpyAsync(buf0, buf1, (size_t)N1 * D * 4, hipMemcpyDeviceToDevice, stream);
        agg_edges_kernel<<<E, 256, 0, stream>>>(buf1, buf0, src2, dst2, valid2, E, D);
        launch_gemm(buf0, g2W, g2b, buf1, N1, D, D, 0, stream);
        bn_stats_kernel<<<D, 256, 0, stream>>>(buf1, N1, D, mu, rstd);
        {
            long long tot = (long long)N1 * D;
            bn_apply_kernel<<<(unsigned)((tot + 255) / 256), 256, 0, stream>>>(buf1, tot, D, mu, rstd, g2g, g2bt);
        }

        // ---- TopK pool 2 (per-graph n = k1, pad to p2pow) ----
        wnorm_kernel<<<1, 256, 0, stream>>>(p2w, D, invn);
        score_kernel<<<(N1 * 32 + 255) / 256, 256, 0, stream>>>(buf1, N1, D, p2w, invn, scores);
        topk_kernel<<<B, 256, (size_t)p2pow * 8, stream>>>(scores, k1, p2pow, k2, perm, vals, mapping);
        int N2 = B * k2;
        gather_kernel<<<N2, 256, 0, stream>>>(buf1, buf0, perm, vals, N2, D);
        // (edges not needed beyond this point)

        // ---- gated attention aggregation over buf0 [B*k2, 512] ----
        launch_gemm(buf0, Pm[28], Pm[29], abuf, N2, D, 256, 2, stream);   // tanh(a)
        launch_gemm(buf0, Pm[30], Pm[31], bbuf, N2, D, 256, 3, stream);   // sigmoid(b)
        gate_kernel<<<(N2 * 32 + 255) / 256, 256, 0, stream>>>(abuf, bbuf, Pm[32], Pm[33], N2, 256, gateS);
        softmax_kernel<<<B, 256, 0, stream>>>(gateS, k2, wgt);
        pool_agg_kernel<<<dim3(B, 2), 256, 0, stream>>>(buf0, wgt, k2, D, pooled);
    };

    // r branch: params 4..15
    run_branch(x_r, DF_R, edge_r, E_R, NPG_R, K_R1, K_R2, 2048,
               Pm[4], Pm[5], Pm[6], Pm[7], Pm[8], Pm[9], Pm[10],
               Pm[11], Pm[12], Pm[13], Pm[14], Pm[15], pooled_r);
    // p branch: params 16..27
    run_branch(x_p, DF_P, edge_p, E_P, NPG_P, K_P1, K_P2, 2048,
               Pm[16], Pm[17], Pm[18], Pm[19], Pm[20], Pm[21], Pm[22],
               Pm[23], Pm[24], Pm[25], Pm[26], Pm[27], pooled_p);

    // classifier: out[B,4] = concat(pooled_r, pooled_p) @ cls_W + cls_b
    cls_kernel<<<1, 16, 0, stream>>>(pooled_r, pooled_p, Pm[34], Pm[35], (float*)d_out);
}